// noisy_top_k_router_56650618634404
// MI455X (gfx1250) — compile-verified
//
#include <hip/hip_runtime.h>
#include <hip/hip_bf16.h>
#include <math.h>

// Noisy top-k router, fused single pass over mh_out (memory-bound: 96MB @ 23.3TB/s ~ 4.1us floor).
// GEMM via V_WMMA_F32_16X16X4_F32: C[16 tokens x 16 outputs], cols 0-7 = logits (W),
// cols 8-15 = noise linear (Wn). Each tile's K=768 reduction is split across a PAIR of
// waves (latency hiding: 4096 waves total), partials reduced through LDS.
// Epilogue: noisy = logit + noise*softplus(nl), top-2, softmax.

typedef __attribute__((ext_vector_type(2))) float v2f;
typedef __attribute__((ext_vector_type(8))) float v8f;

#define E_DIM 768
#define NE_DIM 8
#define KSPLIT 2
#define K_PER_WAVE (E_DIM / KSPLIT)        // 384
#define LDSW_STRIDE 772                    // 772 % 64 == 4 -> conflict-free b64 B reads
#define WAVES_PER_BLOCK 8
#define TILES_PER_BLOCK (WAVES_PER_BLOCK / KSPLIT)             // 4
#define TOKENS_PER_WAVE 16
#define TOKENS_PER_BLOCK (TILES_PER_BLOCK * TOKENS_PER_WAVE)   // 64

__launch_bounds__(256)
__global__ void noisy_topk_router_kernel(const float* __restrict__ mh,    // [B*S, E]
                                         const float* __restrict__ W,     // [NE, E]
                                         const float* __restrict__ b,     // [NE]
                                         const float* __restrict__ Wn,    // [NE, E]
                                         const float* __restrict__ bn,    // [NE]
                                         const float* __restrict__ noise, // [B*S, NE]
                                         float* __restrict__ ro,          // [B*S, NE]
                                         float* __restrict__ ind)         // [B*S, 2] (float-encoded)
{
    __shared__ float ldsW[16 * LDSW_STRIDE];                    // stacked [W; Wn], padded rows
    __shared__ float ldsC[WAVES_PER_BLOCK][TOKENS_PER_WAVE][17];// padded partial-C scratch

    const int tid = threadIdx.x;

    // Stage combined 16x768 weight matrix into LDS (coalesced over the block).
    for (int i = tid; i < 16 * E_DIM; i += 256) {
        const int n = i / E_DIM;
        const int k = i - n * E_DIM;
        const float w = (n < NE_DIM) ? W[n * E_DIM + k] : Wn[(n - NE_DIM) * E_DIM + k];
        ldsW[n * LDSW_STRIDE + k] = w;
    }
    __syncthreads();

    const int wave = tid >> 5;   // wave32
    const int lane = tid & 31;
    const int hi   = lane >> 4;  // 0: lanes 0-15, 1: lanes 16-31
    const int l16  = lane & 15;

    const int pair  = wave >> 1;           // which 16-token tile in this block
    const int kh    = wave & 1;            // which K half this wave reduces
    const int tok0  = blockIdx.x * TOKENS_PER_BLOCK + pair * TOKENS_PER_WAVE;
    const int kbase = kh * K_PER_WAVE;

    // C-matrix column owned by this lane is n = l16 (same for all 8 accumulator rows):
    // bias folds into the accumulator init of the K-low wave only.
    const float bias = (l16 < NE_DIM) ? b[l16] : bn[l16 - NE_DIM];
    v8f c0, c1;
#pragma unroll
    for (int v = 0; v < 8; ++v) { c0[v] = (kh == 0) ? bias : 0.0f; c1[v] = 0.0f; }

    // A fragment (16x4 f32 layout): lane row M = l16, K pair base = 2*hi. b64 loads.
    const float* arow = mh + (size_t)(tok0 + l16) * E_DIM + kbase + 2 * hi;
    // B fragment (4x16 layout): lane col N = l16, K pair base = 2*hi. b64 LDS loads.
    const float* brow = ldsW + l16 * LDSW_STRIDE + kbase + 2 * hi;

    // 64-float (256B/row) chunks: 16 b64 loads + 16 WMMA per chunk. Prefetch 2 chunks
    // ahead at WGP scope (locality 3 -> pulls into all cache levels, so demand b64s hit
    // WGP$). WGP-scope prefetch is non-speculative, so guard it in-bounds; the K loop is
    // fully unrolled (K_PER_WAVE literal) so the guard folds away at compile time.
    for (int k0 = 0; k0 < K_PER_WAVE; k0 += 64) {
        if (k0 + 192 <= K_PER_WAVE) {
            __builtin_prefetch(arow + k0 + 128, 0, 3);
            __builtin_prefetch(arow + k0 + 160, 0, 3);
        }
#pragma unroll
        for (int kk = 0; kk < 64; kk += 8) {
            v2f a0 = *(const v2f*)(arow + k0 + kk);
            v2f b0 = *(const v2f*)(brow + k0 + kk);
            c0 = __builtin_amdgcn_wmma_f32_16x16x4_f32(false, a0, false, b0,
                                                       (short)0, c0, false, false);
            v2f a1 = *(const v2f*)(arow + k0 + kk + 4);
            v2f b1 = *(const v2f*)(brow + k0 + kk + 4);
            c1 = __builtin_amdgcn_wmma_f32_16x16x4_f32(false, a1, false, b1,
                                                       (short)0, c1, false, false);
        }
    }

    // C/D layout: vgpr v, lanes 0-15 -> token M=v, lanes 16-31 -> token M=v+8; N = l16.
#pragma unroll
    for (int v = 0; v < 8; ++v) {
        ldsC[wave][v + 8 * hi][l16] = c0[v] + c1[v];
    }
    __syncthreads();

    // Epilogue: K-low wave of each pair, lanes 0-15, one token per lane.
    if (kh == 0 && hi == 0) {
        const int tok = tok0 + l16;
        float nz[NE_DIM];
#pragma unroll
        for (int n = 0; n < NE_DIM; ++n) {
            const float logit = ldsC[2 * pair][l16][n]          + ldsC[2 * pair + 1][l16][n];
            const float nl    = ldsC[2 * pair][l16][n + NE_DIM] + ldsC[2 * pair + 1][l16][n + NE_DIM];
            // numerically-stable softplus
            const float sp = fmaxf(nl, 0.0f) + log1pf(__expf(-fabsf(nl)));
            nz[n] = logit + noise[(size_t)tok * NE_DIM + n] * sp;
        }
        // top-2 over 8, ties resolve to lowest index first (lax.top_k semantics)
        int i1 = 0; float v1 = nz[0];
#pragma unroll
        for (int n = 1; n < NE_DIM; ++n) { if (nz[n] > v1) { v1 = nz[n]; i1 = n; } }
        int i2 = -1; float v2 = -INFINITY;
#pragma unroll
        for (int n = 0; n < NE_DIM; ++n) { if (n != i1 && nz[n] > v2) { v2 = nz[n]; i2 = n; } }

        // softmax over {v1, v2}; all other entries exactly 0 (exp(-inf))
        const float e2  = __expf(v2 - v1);
        const float inv = 1.0f / (1.0f + e2);
#pragma unroll
        for (int n = 0; n < NE_DIM; ++n) {
            const float r = (n == i1) ? inv : ((n == i2) ? e2 * inv : 0.0f);
            ro[(size_t)tok * NE_DIM + n] = r;
        }
        ind[(size_t)tok * 2 + 0] = (float)i1;
        ind[(size_t)tok * 2 + 1] = (float)i2;
    }
}

extern "C" void kernel_launch(void* const* d_in, const int* in_sizes, int n_in,
                              void* d_out, int out_size, void* d_ws, size_t ws_size,
                              hipStream_t stream) {
    const float* mh    = (const float*)d_in[0];
    const float* W     = (const float*)d_in[1];
    const float* b     = (const float*)d_in[2];
    const float* Wn    = (const float*)d_in[3];
    const float* bn    = (const float*)d_in[4];
    const float* noise = (const float*)d_in[5];
    // d_in[6] is top_k (==2), compile-time constant here.

    const int tokens = in_sizes[0] / E_DIM;   // B*S = 32768
    float* ro  = (float*)d_out;
    float* ind = ro + (size_t)tokens * NE_DIM;

    dim3 grid(tokens / TOKENS_PER_BLOCK);     // 512 blocks x 8 waves = 4096 waves (2 per tile)
    noisy_topk_router_kernel<<<grid, 256, 0, stream>>>(mh, W, b, Wn, bn, noise, ro, ind);
}